// COSYNN_18975165514289
// MI455X (gfx1250) — compile-verified
//
#include <hip/hip_runtime.h>
#include <hip/hip_bf16.h>

#define KAPPA 8
#define TEMPF 200.0f
#define EPSV  1e-15f
#define NN    8192
#define EE    131072
#define DD    128
#define TT    8
#define PP0   512
#define PP1   128
#define ZZI   120

typedef __attribute__((ext_vector_type(2))) float v2f;
typedef __attribute__((ext_vector_type(8))) float v8f;

__global__ void fill_zero_kernel(float* __restrict__ p, long n) {
  long i = (long)blockIdx.x * blockDim.x + threadIdx.x;
  if (i < n) p[i] = 0.0f;
}

__global__ void copy_kernel(const float* __restrict__ s, float* __restrict__ d, long n) {
  long i = (long)blockIdx.x * blockDim.x + threadIdx.x;
  if (i < n) d[i] = s[i];
}

// Zero-pad ZZIxZZI weight + bias into 128x128 / 128
__global__ void pad_we_kernel(const float* __restrict__ W, const float* __restrict__ b,
                              float* __restrict__ Wp, float* __restrict__ bp) {
  int i = blockIdx.x * blockDim.x + threadIdx.x;   // 0..16383
  int r = i >> 7, c = i & 127;
  Wp[i] = (r < ZZI && c < ZZI) ? W[r * ZZI + c] : 0.0f;
  if (r == 0) bp[c] = (c < ZZI) ? b[c] : 0.0f;
}

// ---- GEMM helpers -----------------------------------------------------------

// Stage one 64x32 A tile + 32x32 B tile (B transposed to [n][k]) into LDS.
// GUARD=false: K-tile fully in range (branch-free fast path + prefetch).
template <bool TA, bool GUARD>
__device__ __forceinline__ void stage_tiles(const float* __restrict__ A, long lda,
                                            const float* __restrict__ B, long ldb,
                                            int tm, int tn, int k0, int K, int t,
                                            float* __restrict__ Ash, float* __restrict__ Bsh) {
  if (!TA) {
    const int r  = t >> 2;                 // 0..63
    const int cq = (t & 3) * 8;            // 0,8,16,24
    const float* ga = A + (long)(tm + r) * lda + k0 + cq;
    float4 a0, a1;
    if (!GUARD) {
      a0 = *(const float4*)ga;
      a1 = *(const float4*)(ga + 4);
      __builtin_prefetch(ga + 32, 0, 0);   // global_prefetch_b8 (non-faulting)
    } else {
      float tmp[8];
#pragma unroll
      for (int j = 0; j < 8; ++j) tmp[j] = (k0 + cq + j < K) ? ga[j] : 0.0f;
      a0 = make_float4(tmp[0], tmp[1], tmp[2], tmp[3]);
      a1 = make_float4(tmp[4], tmp[5], tmp[6], tmp[7]);
    }
    *(float4*)&Ash[r * 36 + cq]     = a0;
    *(float4*)&Ash[r * 36 + cq + 4] = a1;
  } else {
    const int k  = t >> 3;                 // 0..31
    const int m0 = (t & 7) * 8;            // 0..56  (m-contiguous in memory)
    const float* ga = A + (long)(k0 + k) * lda + tm + m0;
    float4 a0, a1;
    if (!GUARD) {
      a0 = *(const float4*)ga;
      a1 = *(const float4*)(ga + 4);
      __builtin_prefetch(ga + 32 * lda, 0, 0);
    } else if (k0 + k < K) {
      a0 = *(const float4*)ga;
      a1 = *(const float4*)(ga + 4);
    } else {
      a0 = make_float4(0.f, 0.f, 0.f, 0.f); a1 = a0;
    }
    Ash[(m0 + 0) * 36 + k] = a0.x; Ash[(m0 + 1) * 36 + k] = a0.y;
    Ash[(m0 + 2) * 36 + k] = a0.z; Ash[(m0 + 3) * 36 + k] = a0.w;
    Ash[(m0 + 4) * 36 + k] = a1.x; Ash[(m0 + 5) * 36 + k] = a1.y;
    Ash[(m0 + 6) * 36 + k] = a1.z; Ash[(m0 + 7) * 36 + k] = a1.w;
  }
  {
    const int k  = t >> 3;                 // 0..31
    const int n0 = (t & 7) * 4;            // 0..28
    const float* gb = B + (long)(k0 + k) * ldb + tn + n0;
    float4 b0;
    if (!GUARD) {
      b0 = *(const float4*)gb;
      __builtin_prefetch(gb + 32 * ldb, 0, 0);
    } else if (k0 + k < K) {
      b0 = *(const float4*)gb;
    } else {
      b0 = make_float4(0.f, 0.f, 0.f, 0.f);
    }
    Bsh[(n0 + 0) * 36 + k] = b0.x; Bsh[(n0 + 1) * 36 + k] = b0.y;
    Bsh[(n0 + 2) * 36 + k] = b0.z; Bsh[(n0 + 3) * 36 + k] = b0.w;
  }
}

// 8 back-to-back f32 WMMAs over one staged K-tile (ds_load_b64 feeds).
__device__ __forceinline__ void wmma_ktile(const float* __restrict__ ap,
                                           const float* __restrict__ bp, v8f& acc) {
#pragma unroll
  for (int ks = 0; ks < 32; ks += 4) {
    v2f a = *(const v2f*)(ap + ks);
    v2f b = *(const v2f*)(bp + ks);
    acc = __builtin_amdgcn_wmma_f32_16x16x4_f32(false, a, false, b, (short)0, acc,
                                                false, false);
  }
}

// Tiled f32 WMMA GEMM. Block = 256 threads (8 waves) -> 64(M) x 32(N) tile, Kt=32.
// TA=false: C = A(MxK, row-major lda) * B(KxN, row-major ldb)
// TA=true : C = A^T * B where A is KxM row-major (lda)
// Requires M % 64 == 0, N % 32 == 0; any K (tail zero-padded in LDS).
template <bool TA>
__global__ __launch_bounds__(256)
void gemm_wmma_tiled(const float* __restrict__ A, long lda,
                     const float* __restrict__ B, long ldb,
                     float* __restrict__ C, long ldc,
                     const float* __restrict__ bias, int K) {
  __shared__ float Ash[64 * 36];   // [m][k], k-contiguous, row stride 36 (8B-aligned pairs)
  __shared__ float Bsh[32 * 36];   // [n][k], transposed at stage time
  const int t    = threadIdx.x;    // 0..255
  const int lane = t & 31;
  const int wave = t >> 5;         // 0..7
  const int wm   = wave >> 1;      // 0..3
  const int wn   = wave & 1;       // 0..1
  const int half = lane >> 4;      // f32 WMMA K-phase select (ISA 7.12.2)
  const int l16  = lane & 15;
  const int tm   = blockIdx.y * 64;
  const int tn   = blockIdx.x * 32;

  const float* ap = &Ash[(wm * 16 + l16) * 36 + 2 * half];
  const float* bp = &Bsh[(wn * 16 + l16) * 36 + 2 * half];

  v8f acc = {};
  const int kfull = K & ~31;
  for (int k0 = 0; k0 < kfull; k0 += 32) {      // branch-free steady state
    stage_tiles<TA, false>(A, lda, B, ldb, tm, tn, k0, K, t, Ash, Bsh);
    __syncthreads();
    wmma_ktile(ap, bp, acc);
    __syncthreads();
  }
  if (kfull < K) {                               // single guarded tail tile (K=120 cases)
    stage_tiles<TA, true>(A, lda, B, ldb, tm, tn, kfull, K, t, Ash, Bsh);
    __syncthreads();
    wmma_ktile(ap, bp, acc);
  }

  const int cn = tn + wn * 16 + l16;
  const float bv = bias ? bias[cn] : 0.0f;
#pragma unroll
  for (int r = 0; r < 8; ++r)
    C[(long)(tm + wm * 16 + r + 8 * half) * ldc + cn] = acc[r] + bv;
}

// ---- graph / nonlinearity kernels ------------------------------------------

// dst[dsti[e], :] += src[srci[e], :]  over W columns (segment_sum / A0 @ X)
__global__ void scatter_add_kernel(const int* __restrict__ dsti, const int* __restrict__ srci,
                                   const float* __restrict__ src, float* __restrict__ dst, int W) {
  long i = (long)blockIdx.x * blockDim.x + threadIdx.x;
  long tot = (long)EE * W;
  if (i >= tot) return;
  int e = (int)(i / W);
  int c = (int)(i % W);
  atomicAdd(&dst[(long)dsti[e] * W + c], src[(long)srci[e] * W + c]);
}

// In-place: u <- logmap0(expmap0(u)) * TEMP  == u * atanh(clip(tanh(|u|),eps,1-1e-5))/|u| * TEMP
__global__ void row_logits_kernel(float* __restrict__ U, int p) {
  __shared__ float red[256];
  float* u = U + (long)blockIdx.x * p;
  float ss = 0.f;
  for (int c = threadIdx.x; c < p; c += blockDim.x) { float v = u[c]; ss += v * v; }
  red[threadIdx.x] = ss; __syncthreads();
  for (int s = blockDim.x >> 1; s > 0; s >>= 1) {
    if ((int)threadIdx.x < s) red[threadIdx.x] += red[threadIdx.x + s];
    __syncthreads();
  }
  float nu = fmaxf(sqrtf(red[0]), EPSV);
  float t  = fminf(fmaxf(tanhf(nu), EPSV), 1.0f - 1e-5f);
  float f  = atanhf(t) / nu * TEMPF;
  for (int c = threadIdx.x; c < p; c += blockDim.x) u[c] *= f;
}

// Column-wise softmax (axis 0) of L[n x p] -> S, accumulate sum(entr(S)) into *loss.
__global__ void col_softmax_ent_kernel(const float* __restrict__ L, float* __restrict__ S,
                                       int n, int p, float* __restrict__ loss) {
  __shared__ float red[256];
  const int col = blockIdx.x;
  const int t = threadIdx.x;
  float m = -3.4e38f;
  for (int i = t; i < n; i += blockDim.x) m = fmaxf(m, L[(long)i * p + col]);
  red[t] = m; __syncthreads();
  for (int s = blockDim.x >> 1; s > 0; s >>= 1) { if (t < s) red[t] = fmaxf(red[t], red[t + s]); __syncthreads(); }
  m = red[0]; __syncthreads();
  float sum = 0.f;
  for (int i = t; i < n; i += blockDim.x) sum += expf(L[(long)i * p + col] - m);
  red[t] = sum; __syncthreads();
  for (int s = blockDim.x >> 1; s > 0; s >>= 1) { if (t < s) red[t] += red[t + s]; __syncthreads(); }
  sum = red[0]; __syncthreads();
  const float inv = 1.0f / sum;
  float ent = 0.f;
  for (int i = t; i < n; i += blockDim.x) {
    float pr = expf(L[(long)i * p + col] - m) * inv;
    S[(long)i * p + col] = pr;
    ent += (pr > 0.f) ? -pr * logf(pr) : 0.f;
  }
  red[t] = ent; __syncthreads();
  for (int s = blockDim.x >> 1; s > 0; s >>= 1) { if (t < s) red[t] += red[t + s]; __syncthreads(); }
  if (t == 0) atomicAdd(loss, red[0]);
}

// z[:, :8] = x[:, :8]; z[:, 8:128] = zi[:, 0:120]   (zi stored width 128, cols >=120 are zero)
__global__ void concat_z_kernel(const float* __restrict__ xfull, const float* __restrict__ zi,
                                float* __restrict__ z, int n) {
  long i = (long)blockIdx.x * blockDim.x + threadIdx.x;
  if (i >= (long)n * DD) return;
  int c = (int)(i & (DD - 1));
  z[i] = (c < KAPPA) ? xfull[i] : zi[i - KAPPA];
}

// copies TT x W into dst with row stride dstStride
__global__ void copy_y_kernel(const float* __restrict__ y, float* __restrict__ dst,
                              int W, int dstStride) {
  long i = (long)blockIdx.x * blockDim.x + threadIdx.x;
  if (i >= (long)TT * W) return;
  int tt = (int)(i / W), c = (int)(i % W);
  dst[(long)tt * dstStride + c] = y[i];
}

// out[t,p] = sum_i Y[t,i] * S[i,p];  optional compact copy dA (TT x P), strided dB
__global__ void y_project_kernel(const float* __restrict__ Y, int K,
                                 const float* __restrict__ S, int P,
                                 float* __restrict__ dA, float* __restrict__ dB, int strideB) {
  int idx = blockIdx.x * blockDim.x + threadIdx.x;
  if (idx >= TT * P) return;
  int tt = idx / P, pp = idx % P;
  const float* yr = Y + (long)tt * K;
  float acc = 0.f;
  for (int i = 0; i < K; ++i) acc += yr[i] * S[(long)i * P + pp];
  if (dA) dA[idx] = acc;
  dB[(long)tt * strideB + pp] = acc;
}

static inline unsigned cdiv(long a, long b) { return (unsigned)((a + b - 1) / b); }

extern "C" void kernel_launch(void* const* d_in, const int* in_sizes, int n_in,
                              void* d_out, int out_size, void* d_ws, size_t ws_size,
                              hipStream_t stream) {
  (void)in_sizes; (void)n_in; (void)out_size; (void)ws_size;
  const float* x   = (const float*)d_in[0];
  const float* y   = (const float*)d_in[1];
  const float* We0 = (const float*)d_in[2];
  const float* be0 = (const float*)d_in[3];
  const float* Ws0 = (const float*)d_in[4];
  const float* bs0 = (const float*)d_in[5];
  const float* We1 = (const float*)d_in[6];
  const float* be1 = (const float*)d_in[7];
  const float* Ws1 = (const float*)d_in[8];
  const float* bs1 = (const float*)d_in[9];
  const int*   adj = (const int*)d_in[10];
  const int* adj0 = adj;
  const int* adj1 = adj + EE;

  float* ws   = (float*)d_ws;
  float* We0p = ws;                       // 128x128
  float* be0p = We0p + 16384;             // 128
  float* We1p = be0p + 128;               // 128x128
  float* be1p = We1p + 16384;             // 128
  float* h0   = be1p + 128;               // N x 128
  float* g0   = h0   + (long)NN * DD;     // N x 512  (later aliased as T0)
  float* sagg = g0   + (long)NN * PP0;    // N x 512
  float* ziag = sagg + (long)NN * PP0;    // N x 128
  float* S0   = ziag + (long)NN * DD;     // N x 512
  float* zbuf = S0   + (long)NN * PP0;    // N x 128
  float* A1   = zbuf + (long)NN * DD;     // 512 x 512
  float* h1   = A1   + PP0 * PP0;         // 512 x 128
  float* zi1  = h1   + PP0 * DD;
  float* g1   = zi1  + PP0 * DD;
  float* u1   = g1   + PP0 * DD;
  float* S1   = u1   + PP0 * DD;
  float* z1   = S1   + PP0 * DD;
  float* y1s  = z1   + PP0 * DD;          // 8 x 512
  float* T0   = g0;                       // alias: g0 dead after scatter

  float* out  = (float*)d_out;
  float* zr   = out;                              // (8192+512+128) x 128
  float* x1   = zr + (long)NN * DD;               // rows 8192..8703
  float* x2   = x1 + (long)PP0 * DD;              // rows 8704..8831
  const int YRW = NN + PP0 + PP1;                 // 8832
  float* yr   = zr + (long)YRW * DD;              // 8 x 8832
  float* loss = yr + (long)TT * YRW;              // scalar

  // ---- weight padding, loss init ----
  pad_we_kernel<<<64, 256, 0, stream>>>(We0, be0, We0p, be0p);
  pad_we_kernel<<<64, 256, 0, stream>>>(We1, be1, We1p, be1p);
  fill_zero_kernel<<<1, 32, 0, stream>>>(loss, 1);

  // ---- level 0 dense GEMMs (tiled f32 WMMA) ----
  // h0 = x[:,8:] @ We0p + be0p   (M=8192, N=128, K=120)
  gemm_wmma_tiled<false><<<dim3(DD / 32, NN / 64), 256, 0, stream>>>(x + KAPPA, DD, We0p, DD, h0, DD, be0p, ZZI);
  // g0 = x @ Ws0 + bs0           (M=8192, N=512, K=128)
  gemm_wmma_tiled<false><<<dim3(PP0 / 32, NN / 64), 256, 0, stream>>>(x, DD, Ws0, PP0, g0, PP0, bs0, DD);

  // ---- agg0: scatter-add over edges ----
  fill_zero_kernel<<<cdiv((long)NN * DD, 256), 256, 0, stream>>>(ziag, (long)NN * DD);
  fill_zero_kernel<<<cdiv((long)NN * PP0, 256), 256, 0, stream>>>(sagg, (long)NN * PP0);
  scatter_add_kernel<<<cdiv((long)EE * DD, 256), 256, 0, stream>>>(adj0, adj1, h0, ziag, DD);
  scatter_add_kernel<<<cdiv((long)EE * PP0, 256), 256, 0, stream>>>(adj0, adj1, g0, sagg, PP0);

  // ---- logits = logmap0(expmap0(.))*TEMP, column softmax -> S0 (+entropy) ----
  row_logits_kernel<<<NN, 256, 0, stream>>>(sagg, PP0);
  col_softmax_ent_kernel<<<PP0, 256, 0, stream>>>(sagg, S0, NN, PP0, loss);

  // z = concat(x[:,:8], ziag[:,:120])
  concat_z_kernel<<<cdiv((long)NN * DD, 256), 256, 0, stream>>>(x, ziag, zbuf, NN);

  // T0 = A0 @ S0 (sparse scatter), then A1 = S0^T T0 (dominant GEMM, K=8192)
  fill_zero_kernel<<<cdiv((long)NN * PP0, 256), 256, 0, stream>>>(T0, (long)NN * PP0);
  scatter_add_kernel<<<cdiv((long)EE * PP0, 256), 256, 0, stream>>>(adj0, adj1, S0, T0, PP0);
  gemm_wmma_tiled<true><<<dim3(PP0 / 32, PP0 / 64), 256, 0, stream>>>(S0, PP0, T0, PP0, A1, PP0, nullptr, NN);

  // x1 = S0^T z  -> straight into z_r rows 8192..8703
  gemm_wmma_tiled<true><<<dim3(DD / 32, PP0 / 64), 256, 0, stream>>>(S0, PP0, zbuf, DD, x1, DD, nullptr, NN);
  // y1 = y @ S0  -> scratch + y_r columns 8192..8703
  y_project_kernel<<<cdiv(TT * PP0, 256), 256, 0, stream>>>(y, NN, S0, PP0, y1s, yr + NN, YRW);

  // ---- level 1 (n=512, agg = A1 @ .) ----
  gemm_wmma_tiled<false><<<dim3(DD / 32, PP0 / 64), 256, 0, stream>>>(x1 + KAPPA, DD, We1p, DD, h1, DD, be1p, ZZI);
  gemm_wmma_tiled<false><<<dim3(DD / 32, PP0 / 64), 256, 0, stream>>>(A1, PP0, h1, DD, zi1, DD, nullptr, PP0);
  gemm_wmma_tiled<false><<<dim3(PP1 / 32, PP0 / 64), 256, 0, stream>>>(x1, DD, Ws1, PP1, g1, PP1, bs1, DD);
  gemm_wmma_tiled<false><<<dim3(PP1 / 32, PP0 / 64), 256, 0, stream>>>(A1, PP0, g1, PP1, u1, PP1, nullptr, PP0);

  row_logits_kernel<<<PP0, 256, 0, stream>>>(u1, PP1);
  col_softmax_ent_kernel<<<PP1, 256, 0, stream>>>(u1, S1, PP0, PP1, loss);

  concat_z_kernel<<<cdiv((long)PP0 * DD, 256), 256, 0, stream>>>(x1, zi1, z1, PP0);

  // x2 = S1^T z1 -> z_r rows 8704..8831
  gemm_wmma_tiled<true><<<dim3(DD / 32, PP1 / 64), 256, 0, stream>>>(S1, PP1, z1, DD, x2, DD, nullptr, PP0);
  // y2 = y1 @ S1 -> y_r columns 8704..8831
  y_project_kernel<<<cdiv(TT * PP1, 256), 256, 0, stream>>>(y1s, PP0, S1, PP1, nullptr, yr + NN + PP0, YRW);

  // ---- assemble passthrough pieces of outputs ----
  copy_kernel<<<cdiv((long)NN * DD, 256), 256, 0, stream>>>(x, zr, (long)NN * DD);
  copy_y_kernel<<<cdiv((long)TT * NN, 256), 256, 0, stream>>>(y, yr, NN, YRW);
}